// MultiScaleROIPool_55052890800547
// MI455X (gfx1250) — compile-verified
//
#include <hip/hip_runtime.h>
#include <stdint.h>

// Multi-scale ROI align (FPN), fused level selection, CDNA5 async-to-LDS gather,
// double-buffered on ASYNCcnt with partial-counter waits.
// B=4, C=256, N=256, OUT=7, SR=2. Output [B*N, C, 7, 7] f32.

#define NTAP     28          // 14 samples/axis * 2 bilinear taps
#define PSTRIDE  29          // +1 padding to spread LDS banks
#define CHUNK    16          // channels per chunk: CHUNK*784 == 49*256 -> exactly 49 async issues per wave
#define NTHREADS 256
#define ISSUES_PER_WAVE 49   // per chunk, uniform across all 8 waves

// ---- CDNA5 async global->LDS gather (per-lane scattered 4B, tracked on ASYNCcnt) ----
__device__ __forceinline__ void async_copy_b32(const float* g, float* l) {
#if __has_builtin(__builtin_amdgcn_global_load_async_to_lds_b32)
  __builtin_amdgcn_global_load_async_to_lds_b32((int*)g, (int*)l, 0, 0);
#else
  *l = *g;
#endif
}

// Wait until at most `n` async-loads are still outstanding for this wave.
// Async loads complete in order (ISA 08 §4.1), so waiting to ISSUES_PER_WAVE
// after issuing the next chunk guarantees the previous chunk has fully landed.
__device__ __forceinline__ void wait_async_prev() {
#if __has_builtin(__builtin_amdgcn_s_wait_asynccnt)
  __builtin_amdgcn_s_wait_asynccnt(ISSUES_PER_WAVE);
#else
  asm volatile("" ::: "memory");
#endif
}
__device__ __forceinline__ void wait_async_all() {
#if __has_builtin(__builtin_amdgcn_s_wait_asynccnt)
  __builtin_amdgcn_s_wait_asynccnt(0);
#else
  asm volatile("" ::: "memory");
#endif
}

__device__ __forceinline__ void issue_chunk(const float* cb, int HW,
                                            const int* offs, float* buf, int tid) {
  // 12544 scattered 4B gathers; lanes walk q fastest -> near-contiguous global
  // addresses and contiguous LDS destinations. Exactly 49 wave-wide async issues.
  for (int t = tid; t < CHUNK * NTAP * NTAP; t += NTHREADS) {
    const int cc = t / (NTAP * NTAP);
    const int rq = t - cc * (NTAP * NTAP);
    float* lp = buf + cc * (NTAP * PSTRIDE) + (rq / NTAP) * PSTRIDE + (rq % NTAP);
    async_copy_b32(cb + (size_t)cc * HW + offs[rq], lp);
  }
}

__global__ __launch_bounds__(NTHREADS)
void msroi_align_kernel(const float* __restrict__ f0, const float* __restrict__ f1,
                        const float* __restrict__ f2, const float* __restrict__ f3,
                        const float* __restrict__ f4, const float* __restrict__ rois,
                        float* __restrict__ out, int N, int C)
{
  __shared__ float patch[2][CHUNK][NTAP][PSTRIDE];   // ~104 KB, double-buffered
  __shared__ float rW[NTAP], cW[NTAP];
  __shared__ int   rI[NTAP], cI[NTAP];
  __shared__ int   offs[NTAP * NTAP];                // channel-invariant gather offsets

  const int bn  = blockIdx.x;       // roi index in [0, B*N)
  const int b   = bn / N;
  const int tid = threadIdx.x;

  // ---- uniform per-roi setup: FPN level selection ----
  const float4 roi  = reinterpret_cast<const float4*>(rois)[bn];
  const float  area = (roi.z - roi.x) * (roi.w - roi.y);
  float lvlf = floorf(4.0f + log2f(sqrtf(area) / 224.0f + 1e-6f));
  lvlf = fminf(fmaxf(lvlf, 2.0f), 6.0f);
  const int l = (int)lvlf - 2;

  const float* fbase; int H;
  switch (l) {
    case 0:  fbase = f0; H = 128; break;
    case 1:  fbase = f1; H = 64;  break;
    case 2:  fbase = f2; H = 32;  break;
    case 3:  fbase = f3; H = 16;  break;
    default: fbase = f4; H = 8;   break;
  }
  const float scale = 1.0f / (float)(4 << l);
  const float x1 = roi.x * scale, y1 = roi.y * scale;
  const float x2 = roi.z * scale, y2 = roi.w * scale;
  const float bw = fmaxf(x2 - x1, 1.0f) * (1.0f / 7.0f);
  const float bh = fmaxf(y2 - y1, 1.0f) * (1.0f / 7.0f);

  // ---- separable tap tables: tap r = 2*i + a, sample i = 2*oy + s ----
  if (tid < NTAP) {
    const int   r = tid, i = r >> 1, a = r & 1;
    const float g = (float)(i >> 1) + ((float)(i & 1) + 0.5f) * 0.5f;  // 0.25,0.75,...,6.75
    const float Hf = (float)H;
    // y axis (0.5 folds the SR-average; validity folded multiplicatively)
    const float py = y1 + g * bh;
    const float vy = (py >= -1.0f && py <= Hf) ? 0.5f : 0.0f;
    const float y  = fminf(fmaxf(py, 0.0f), Hf - 1.0f);
    const float yf = floorf(y);
    const float ly = y - yf;
    const int   y0 = (int)yf;
    int yi = y0 + a; if (yi > H - 1) yi = H - 1;
    rI[r] = yi;
    rW[r] = vy * (a ? ly : 1.0f - ly);
    // x axis (feature maps are square)
    const float px = x1 + g * bw;
    const float vx = (px >= -1.0f && px <= Hf) ? 0.5f : 0.0f;
    const float x  = fminf(fmaxf(px, 0.0f), Hf - 1.0f);
    const float xf = floorf(x);
    const float lx = x - xf;
    const int   x0 = (int)xf;
    int xi = x0 + a; if (xi > H - 1) xi = H - 1;
    cI[r] = xi;
    cW[r] = vx * (a ? lx : 1.0f - lx);
  }
  __syncthreads();

  for (int t = tid; t < NTAP * NTAP; t += NTHREADS)
    offs[t] = rI[t / NTAP] * H + cI[t % NTAP];
  __syncthreads();

  // ---- software-pipelined channel-chunk loop ----
  const int    HW      = H * H;
  const float* roibase = fbase + (size_t)b * C * HW;
  const int    NCHUNK  = C / CHUNK;

  issue_chunk(roibase, HW, offs, &patch[0][0][0][0], tid);

  for (int k = 0; k < NCHUNK; ++k) {
    if (k + 1 < NCHUNK) {
      // prefetch next chunk into the other buffer, then wait only for chunk k
      issue_chunk(roibase + (size_t)(k + 1) * CHUNK * HW, HW, offs,
                  &patch[(k + 1) & 1][0][0][0], tid);
      wait_async_prev();     // s_wait_asynccnt 49: chunk k landed, k+1 in flight
    } else {
      wait_async_all();      // s_wait_asynccnt 0
    }
    __syncthreads();

    const float* bufk = &patch[k & 1][0][0][0];
    float* op = out + ((size_t)bn * C + (size_t)k * CHUNK) * 49;
    for (int t = tid; t < CHUNK * 49; t += NTHREADS) {
      const int cc = t / 49;
      const int p  = t - cc * 49;
      const int oy = p / 7, ox = p - oy * 7;
      const float* base = bufk + cc * (NTAP * PSTRIDE) + (4 * oy) * PSTRIDE + 4 * ox;
      float acc = 0.0f;
#pragma unroll
      for (int dr = 0; dr < 4; ++dr) {
        const float* row = base + dr * PSTRIDE;
        const float  s   = cW[4 * ox + 0] * row[0] + cW[4 * ox + 1] * row[1]
                         + cW[4 * ox + 2] * row[2] + cW[4 * ox + 3] * row[3];
        acc = fmaf(rW[4 * oy + dr], s, acc);
      }
      // streaming store: output is write-once, keep it from rinsing feats out of L2
      __builtin_nontemporal_store(acc, op + t);   // t == cc*49 + p: contiguous
    }
    __syncthreads();   // buffer (k&1) free before iteration k+1 reissues into it
  }
}

extern "C" void kernel_launch(void* const* d_in, const int* in_sizes, int n_in,
                              void* d_out, int out_size, void* d_ws, size_t ws_size,
                              hipStream_t stream) {
  const float* f0   = (const float*)d_in[0];
  const float* f1   = (const float*)d_in[1];
  const float* f2   = (const float*)d_in[2];
  const float* f3   = (const float*)d_in[3];
  const float* f4   = (const float*)d_in[4];
  const float* rois = (const float*)d_in[5];
  float*       out  = (float*)d_out;

  const int BN = in_sizes[5] / 4;          // B*N rois
  const int N  = 256;                      // per reference
  const int B  = BN / N;
  const int C  = in_sizes[0] / (B * 128 * 128);

  msroi_align_kernel<<<dim3(BN), dim3(NTHREADS), 0, stream>>>(
      f0, f1, f2, f3, f4, rois, out, N, C);
}